// ChamferDistanceLoss_86792699118040
// MI455X (gfx1250) — compile-verified
//
#include <hip/hip_runtime.h>
#include <math.h>

// Chamfer distance, B=8, N=M=4096, D=3, fp32.
// dist^2(x,y) = |x|^2 + |y|^2 - 2 x.y. The affine form is baked directly into
// V_WMMA_F32_16X16X4_F32:  D = (-2*X) x Y^T + (|x|^2 + |y|^2), so dist^2 pops
// out of the matrix op with no per-element FMA fixup. K=3 zero-padded to K=4.
// Row-minima kept in registers; kernel is run twice with inputs swapped to get
// both chamfer directions without any atomics.

typedef __attribute__((ext_vector_type(2))) float v2f;
typedef __attribute__((ext_vector_type(8))) float v8f;

#define NPTS 4096
#define NBATCH 8

// One wave (32 lanes) owns 16 rows of P and scans all 4096 cols of Q.
// Block = 256 threads = 8 waves = 8 row-tiles. Grid = (32 tile-groups, 8 batches).
__global__ __launch_bounds__(256) void chamfer_min_kernel(
    const float* __restrict__ P, const float* __restrict__ Q,
    float* __restrict__ minout)
{
  const int lane  = threadIdx.x & 31;
  const int wave  = threadIdx.x >> 5;
  const int batch = blockIdx.y;
  const int rowTile = blockIdx.x * 8 + wave;   // 0..255
  const int rowBase = rowTile * 16;

  const float* Pb = P + (size_t)batch * NPTS * 3;
  const float* Qb = Q + (size_t)batch * NPTS * 3;

  const int hi = lane >> 4;   // half of the wave
  const int lm = lane & 15;

  // ---- A-matrix (16x4 f32), pre-scaled by -2:
  //      lanes 0-15 carry K=0,1 ; lanes 16-31 carry K=2,3(=0)
  const int myRow = rowBase + lm;
  const float p0 = Pb[myRow * 3 + 0];
  const float p1 = Pb[myRow * 3 + 1];
  const float p2 = Pb[myRow * 3 + 2];
  const float myPn = p0 * p0 + p1 * p1 + p2 * p2;

  v2f amat;
  amat.x = -2.0f * (hi ? p2 : p0);
  amat.y = hi ? 0.0f : (-2.0f * p1);

  // Row-norm table matching the C layout: C VGPR r at half h -> row rowBase+r+8h.
  float pn[8];
#pragma unroll
  for (int r = 0; r < 8; ++r)
    pn[r] = __shfl(myPn, r + 8 * hi, 32);

  float dmin[8];
#pragma unroll
  for (int r = 0; r < 8; ++r) dmin[r] = 3.4e38f;

  for (int ct = 0; ct < NPTS / 16; ++ct) {
    // ---- B-matrix (4x16 f32): N = lm on lanes; lanes 0-15 -> K=0,1 ; 16-31 -> K=2,3(=0)
    const int col = ct * 16 + lm;
    const float q0 = Qb[col * 3 + 0];
    const float q1 = Qb[col * 3 + 1];
    const float q2 = Qb[col * 3 + 2];
    const float qn = q0 * q0 + q1 * q1 + q2 * q2;

    v2f bmat;
    bmat.x = hi ? q2 : q0;
    bmat.y = hi ? 0.0f : q1;

    // C input = |x|^2 + |y|^2 so the WMMA emits dist^2 directly.
    v8f c;
#pragma unroll
    for (int r = 0; r < 8; ++r) c[r] = pn[r] + qn;

    // 8 args: (neg_a, A, neg_b, B, c_mod, C, reuse_a, reuse_b)
    c = __builtin_amdgcn_wmma_f32_16x16x4_f32(
        false, amat, false, bmat, (short)0, c, false, false);

#pragma unroll
    for (int r = 0; r < 8; ++r)
      dmin[r] = fminf(dmin[r], c[r]);
  }

  // Min across the 16 lanes of each half (xor masks 1..8 stay within a half).
#pragma unroll
  for (int r = 0; r < 8; ++r) {
    float v = dmin[r];
#pragma unroll
    for (int off = 1; off < 16; off <<= 1)
      v = fminf(v, __shfl_xor(v, off, 32));
    if (lm == 0) {
      const int row = rowBase + r + 8 * hi;
      minout[(size_t)batch * NPTS + row] = sqrtf(fmaxf(v, 0.0f));
    }
  }
}

// Sum all 2*B*N row-minima; mean_n + mean_m averaged over batches collapses
// to sum/(B*N) since N == M.
__global__ __launch_bounds__(256) void chamfer_reduce_kernel(
    const float* __restrict__ mins, float* __restrict__ out, int total)
{
  __shared__ float sdata[256];
  float s = 0.0f;
  for (int i = threadIdx.x; i < total; i += 256) s += mins[i];
  sdata[threadIdx.x] = s;
  __syncthreads();
  for (int stride = 128; stride > 0; stride >>= 1) {
    if ((int)threadIdx.x < stride) sdata[threadIdx.x] += sdata[threadIdx.x + stride];
    __syncthreads();
  }
  if (threadIdx.x == 0) out[0] = sdata[0] / ((float)NBATCH * (float)NPTS);
}

extern "C" void kernel_launch(void* const* d_in, const int* in_sizes, int n_in,
                              void* d_out, int out_size, void* d_ws, size_t ws_size,
                              hipStream_t stream) {
  const float* x = (const float*)d_in[0];   // [8,4096,3]
  const float* y = (const float*)d_in[1];   // [8,4096,3]
  float* ws   = (float*)d_ws;
  float* minx = ws;                         // [8,4096] min over y for each x
  float* miny = ws + NBATCH * NPTS;         // [8,4096] min over x for each y

  dim3 grid(32, NBATCH);                    // 32 blocks * 8 waves = 256 row tiles
  chamfer_min_kernel<<<grid, 256, 0, stream>>>(x, y, minx);
  chamfer_min_kernel<<<grid, 256, 0, stream>>>(y, x, miny);
  chamfer_reduce_kernel<<<1, 256, 0, stream>>>(ws, (float*)d_out,
                                               2 * NBATCH * NPTS);
}